// Jitter_17849884082575
// MI455X (gfx1250) — compile-verified
//
#include <hip/hip_runtime.h>
#include <stdint.h>

// Jitter gather: out[r, i] = in[r, idx[i]], idx shared across rows, idx[i] in {i-1,i,i+1}.
// B*C = 16384 rows, L = 4096 f32 per row. Pure HBM-bound (512 MB traffic, 0 FLOPs).
// CDNA5 path: async global->LDS b128 staging (non-temporal) + LDS gather +
// coalesced non-temporal b128 stores.

#define L_DIM   4096
#define HALF_L  2048

// Native clang vector types (HIP's float4/int4 are classes, which
// __builtin_nontemporal_store rejects).
typedef float v4f __attribute__((ext_vector_type(4)));
typedef int   v4i __attribute__((ext_vector_type(4)));

// ---------------------------------------------------------------------------
// Threefry-2x32, exactly matching JAX's reference implementation.
// ---------------------------------------------------------------------------
__device__ __forceinline__ uint32_t rotl32(uint32_t x, uint32_t d) {
    return (x << d) | (x >> (32u - d));
}

__device__ __forceinline__ void threefry2x32(uint32_t k0, uint32_t k1,
                                             uint32_t x0, uint32_t x1,
                                             uint32_t& o0, uint32_t& o1) {
    const uint32_t ks0 = k0;
    const uint32_t ks1 = k1;
    const uint32_t ks2 = k0 ^ k1 ^ 0x1BD11BDAu;
    x0 += ks0; x1 += ks1;
#define TF_ROUND(r) { x0 += x1; x1 = rotl32(x1, (r)); x1 ^= x0; }
    TF_ROUND(13) TF_ROUND(15) TF_ROUND(26) TF_ROUND(6)
    x0 += ks1; x1 += ks2 + 1u;
    TF_ROUND(17) TF_ROUND(29) TF_ROUND(16) TF_ROUND(24)
    x0 += ks2; x1 += ks0 + 2u;
    TF_ROUND(13) TF_ROUND(15) TF_ROUND(26) TF_ROUND(6)
    x0 += ks0; x1 += ks1 + 3u;
    TF_ROUND(17) TF_ROUND(29) TF_ROUND(16) TF_ROUND(24)
    x0 += ks1; x1 += ks2 + 4u;
    TF_ROUND(13) TF_ROUND(15) TF_ROUND(26) TF_ROUND(6)
    x0 += ks2; x1 += ks0 + 5u;
#undef TF_ROUND
    o0 = x0; o1 = x1;
}

// JAX uniform in [0,1): bitcast((bits >> 9) | 0x3f800000) - 1.0f
__device__ __forceinline__ float u01(uint32_t b) {
    return __uint_as_float((b >> 9) | 0x3F800000u) - 1.0f;
}

// ---------------------------------------------------------------------------
// Kernel 1: build idx[4096] into workspace. Reproduces:
//   key = (0,42); k_replace,k_dir = split(key); replace = U(k_replace,L)<0.12;
//   dir = U(k_dir,L)<0.5 ? -1 : +1; boundary clamp; idx = replace?pos+dir:pos
// random_bits pairs counter lanes (i, i+2048): first output -> bits[i],
// second output -> bits[i+2048].
// ---------------------------------------------------------------------------
__global__ __launch_bounds__(256) void build_jitter_idx(int* __restrict__ idx) {
    const int i = blockIdx.x * 256 + threadIdx.x;
    if (i >= HALF_L) return;

    // split(key): counts [0,1,2,3] -> lane0=(0,2), lane1=(1,3)
    uint32_t a0, b0, a1, b1;
    threefry2x32(0u, 42u, 0u, 2u, a0, b0);
    threefry2x32(0u, 42u, 1u, 3u, a1, b1);
    // k_replace = (a0, a1) [first outputs], k_dir = (b0, b1) [second outputs]

    uint32_t r0, r1, d0, d1;
    threefry2x32(a0, a1, (uint32_t)i, (uint32_t)(i + HALF_L), r0, r1);
    threefry2x32(b0, b1, (uint32_t)i, (uint32_t)(i + HALF_L), d0, d1);

    {
        const int p = i;
        const bool rep = u01(r0) < 0.12f;
        const int dir  = (u01(d0) < 0.5f) ? -1 : 1;
        const int nb   = (p == 0) ? 1 : ((p == L_DIM - 1) ? (L_DIM - 2) : p + dir);
        idx[p] = rep ? nb : p;
    }
    {
        const int p = i + HALF_L;
        const bool rep = u01(r1) < 0.12f;
        const int dir  = (u01(d1) < 0.5f) ? -1 : 1;
        const int nb   = (p == 0) ? 1 : ((p == L_DIM - 1) ? (L_DIM - 2) : p + dir);
        idx[p] = rep ? nb : p;
    }
}

// ---------------------------------------------------------------------------
// Kernel 2: one block per row. Stage the 16 KB row into LDS with
// global_load_async_to_lds_b128 (ASYNCcnt-tracked CDNA5 async copy, NT hint:
// each input line is touched exactly once device-wide), wait + barrier,
// gather from LDS, store coalesced non-temporal b128 (output never re-read).
// idx loads stay regular-temporal: 16 KB reused by all 16384 blocks -> L2 hit.
// ---------------------------------------------------------------------------
__global__ __launch_bounds__(256) void jitter_gather(const float* __restrict__ in,
                                                     float* __restrict__ out,
                                                     const int* __restrict__ idx) {
    __shared__ float row[L_DIM];

    const int r = blockIdx.x;
    const int t = threadIdx.x;                 // 0..255
    const float* __restrict__ src = in  + (size_t)r * L_DIM;
    float* __restrict__       dst = out + (size_t)r * L_DIM;

    // Low 32 bits of the generic (flat) LDS address == LDS byte offset
    // (flat LDS aperture carries the offset in addr[31:0]).
    const uint32_t lds_base = (uint32_t)(uintptr_t)(&row[0]);

    // Phase A: async-copy the whole row into LDS, fully coalesced b128, NT.
#pragma unroll
    for (int j = 0; j < 4; ++j) {
        const int p = j * 1024 + t * 4;                // float index
        const float* g = src + p;                      // 16B aligned
        const uint32_t l = lds_base + (uint32_t)(p * 4);
        asm volatile("global_load_async_to_lds_b128 %0, %1, off th:TH_LOAD_NT"
                     :
                     : "v"(l), "v"(g)
                     : "memory");
    }
    asm volatile("s_wait_asynccnt 0" ::: "memory");
    __syncthreads();

    // Phase B: gather from LDS and write coalesced non-temporal b128 stores.
#pragma unroll
    for (int j = 0; j < 4; ++j) {
        const int p = j * 1024 + t * 4;
        const v4i ii = *(const v4i*)(idx + p);
        v4f v;
        v.x = row[ii.x];
        v.y = row[ii.y];
        v.z = row[ii.z];
        v.w = row[ii.w];
        __builtin_nontemporal_store(v, (v4f*)(dst + p));
    }
}

// ---------------------------------------------------------------------------
// Launch
// ---------------------------------------------------------------------------
extern "C" void kernel_launch(void* const* d_in, const int* in_sizes, int n_in,
                              void* d_out, int out_size, void* d_ws, size_t ws_size,
                              hipStream_t stream) {
    const float* in = (const float*)d_in[0];
    float* out = (float*)d_out;
    int* idx = (int*)d_ws;                     // 4096 ints = 16 KB scratch

    // 1) Build the shared per-timestep index map (deterministic each call).
    build_jitter_idx<<<HALF_L / 256, 256, 0, stream>>>(idx);

    // 2) Gather: one block per row.
    const int rows = in_sizes[0] / L_DIM;      // 32*512 = 16384
    jitter_gather<<<rows, 256, 0, stream>>>(in, out, idx);
}